// SheafGluingCG_70901320122807
// MI455X (gfx1250) — compile-verified
//
#include <hip/hip_runtime.h>
#include <cstdint>

#define LAM 1.0f
#define TOLV 1e-6f
#define NITERS 20
#define WPB 8  // waves per block (256 threads)

typedef float v2f __attribute__((ext_vector_type(2)));
typedef float v8f __attribute__((ext_vector_type(8)));

__device__ __forceinline__ v8f wmma4(v2f a, v2f b, v8f c) {
  // D = A(16x4 f32) x B(4x16 f32) + C(16x16 f32)
  return __builtin_amdgcn_wmma_f32_16x16x4_f32(false, a, false, b, (short)0, c,
                                               false, false);
}

// ---- CDNA5 async global->LDS path (guarded; falls back to sync staging) ----
#if __has_builtin(__builtin_amdgcn_global_load_async_to_lds_b128)
#define HAVE_ASYNC_LDS 1
#else
#define HAVE_ASYNC_LDS 0
#endif

#if __has_builtin(__builtin_amdgcn_s_wait_asynccnt)
#define WAIT_ASYNC(n) __builtin_amdgcn_s_wait_asynccnt(n)
#else
#define WAIT_ASYNC(n) asm volatile("s_wait_asynccnt %0" ::"i"(n) : "memory")
#endif

#define __AS1 __attribute__((address_space(1)))
#define __AS3 __attribute__((address_space(3)))

#if HAVE_ASYNC_LDS
typedef int vi4 __attribute__((vector_size(16)));  // b128 payload type
__device__ __forceinline__ void async_copy_b128(const void* g, const void* l) {
  // builtin signature: (V4i addrspace(1)*, V4i addrspace(3)*, imm offset, imm cpol)
  // low 32 bits of a generic LDS address are the LDS byte offset
  __builtin_amdgcn_global_load_async_to_lds_b128(
      (__AS1 vi4*)(uintptr_t)g, (__AS3 vi4*)(uint32_t)(uintptr_t)l, 0, 0);
}
#define NBUF 2
#else
#define NBUF 1
#endif

// ---------------------------------------------------------------------------
// Edge kernel: one wave handles 2 edges per trip.
//   stage1: C1 = diag(r0^T, r1^T)       (8 WMMA, K=32)
//   stage2: c_src^T / c_dst^T tiles     (8 WMMA, K=16), B operand = C1
// Async path double-buffers the LDS staging (4 async b128 copies per pair).
// ---------------------------------------------------------------------------
__global__ __launch_bounds__(256) void sheaf_edge_kernel(
    const float* __restrict__ P, const int* __restrict__ srcI,
    const int* __restrict__ dstI, const float* __restrict__ Rs,
    const float* __restrict__ Rd, float* __restrict__ acc, int Mn, int E) {
  __shared__ __align__(16) float Rb[NBUF][WPB][2][128];  // [buf][wave][src/dst][e*64+a*8+d]
  __shared__ __align__(16) float Pb[NBUF][WPB][4][8][8]; // [buf][wave][slice][b][d]

  const int lane = threadIdx.x & 31;
  const int w = threadIdx.x >> 5;
  const int m = lane & 15;   // row/col index inside 16x16 tile
  const int kh = lane >> 4;  // K-half inside a K=4 chunk
  const int lsel = lane & 15;

  const int nPairs = (E + 1) >> 1;
  const int pr0 = blockIdx.x * WPB + w;
  const int stride = gridDim.x * WPB;

#if HAVE_ASYNC_LDS
  auto issueAsync = [&](int pr, int buf) {
    const int e0 = 2 * pr;
    const int e1 = (e0 + 1 < E) ? e0 + 1 : e0;
    // prior DS reads of this buffer must be complete before async overwrite
    asm volatile("s_wait_dscnt 0x0" ::: "memory");
    {
      const size_t ro = (size_t)(kh ? e1 : e0) * 64 + (size_t)lsel * 4;
      async_copy_b128(Rs + ro, &Rb[buf][w][0][lane * 4]);
      async_copy_b128(Rd + ro, &Rb[buf][w][1][lane * 4]);
    }
    {
      const int slice = lane >> 3;
      const int bb = lane & 7;
      const int s0 = srcI[e0], t0 = dstI[e0];
      const int s1 = srcI[e1], t1 = dstI[e1];
      const int node = (slice == 0) ? s0 : (slice == 1) ? t0
                                      : (slice == 2) ? s1 : t1;
      const float* g = P + ((size_t)bb * Mn + node) * 8;
      const float* l = &Pb[buf][w][slice][bb][0];
      async_copy_b128(g, l);
      async_copy_b128(g + 4, l + 4);
    }
  };
#endif

  auto computePair = [&](int pr, int buf) {
    const int e0 = 2 * pr;
    const bool e1v = (e0 + 1) < E;
    const int e1 = e1v ? e0 + 1 : e0;
    const int s0 = srcI[e0], t0 = dstI[e0];
    const int s1 = srcI[e1], t1 = dstI[e1];

    // --- stage 1: C1 = diag(r0^T, r1^T), K = 32 --------------------------
    v8f c1 = {0.f, 0.f, 0.f, 0.f, 0.f, 0.f, 0.f, 0.f};
    const int erow = m >> 3;  // which edge this row/col belongs to
    const int i7 = m & 7;
#pragma unroll
    for (int kc = 0; kc < 8; ++kc) {
      const int k0 = 4 * kc + 2 * kh;
      const float act = (erow == (k0 >> 4)) ? 1.f : 0.f;  // block-diag mask
      const int kk = k0 & 15;
      const int which = kk >> 3;  // 0 = Rs, 1 = -Rd
      const int dd = kk & 7;
      const float sA = which ? -act : act;
      const float* ra = &Rb[buf][w][which][erow * 64 + i7 * 8 + dd];
      v2f av = {sA * ra[0], sA * ra[1]};
      const float* pb = &Pb[buf][w][erow * 2 + which][i7][dd];
      v2f bv = {act * pb[0], act * pb[1]};
      c1 = wmma4(av, bv, c1);
    }

    // --- C-layout -> B-layout: row pairs (j, j+8) -> (4k..4k+3) ----------
#define MLO(a, b)                          \
  ({ float _x = __shfl((a), lsel);         \
     float _y = __shfl((b), lsel);         \
     (lane < 16) ? _x : _y; })
#define MHI(a, b)                          \
  ({ float _x = __shfl((a), lsel + 16);    \
     float _y = __shfl((b), lsel + 16);    \
     (lane < 16) ? _x : _y; })
    v2f b2_0 = {MLO(c1[0], c1[2]), MLO(c1[1], c1[3])};  // rows 0-3
    v2f b2_1 = {MLO(c1[4], c1[6]), MLO(c1[5], c1[7])};  // rows 4-7
    v2f b2_2 = {MHI(c1[0], c1[2]), MHI(c1[1], c1[3])};  // rows 8-11
    v2f b2_3 = {MHI(c1[4], c1[6]), MHI(c1[5], c1[7])};  // rows 12-15
#undef MLO
#undef MHI
    v2f b2[4] = {b2_0, b2_1, b2_2, b2_3};

    // --- stage 2: c_src^T = diag(Rs^T)*C1 ; c_dst^T = diag(-Rd^T)*C1 -----
    v8f cs = {0.f, 0.f, 0.f, 0.f, 0.f, 0.f, 0.f, 0.f};
    v8f cd = {0.f, 0.f, 0.f, 0.f, 0.f, 0.f, 0.f, 0.f};
#pragma unroll
    for (int kc = 0; kc < 4; ++kc) {
      const int k0 = 4 * kc + 2 * kh;  // 0..14
      const float act = (erow == (k0 >> 3)) ? 1.f : 0.f;
      const int kk = k0 & 7;
      const float* rs0 = &Rb[buf][w][0][erow * 64 + kk * 8 + i7];  // Rs[kk,i7]
      const float* rd0 = &Rb[buf][w][1][erow * 64 + kk * 8 + i7];
      v2f as = {act * rs0[0], act * rs0[8]};
      v2f ad = {-act * rd0[0], -act * rd0[8]};
      cs = wmma4(as, b2[kc], cs);
      cd = wmma4(ad, b2[kc], cd);
    }

    // --- scatter: VGPR j = d-index j; lanes 0-7 edge0, 24-31 edge1 -------
    if ((lane < 8) || (lane >= 24)) {
      const int eg = lane >> 4;
      if (!eg || e1v) {
        const int bb = lane & 7;
        const int ns = eg ? s1 : s0;
        const int nt = eg ? t1 : t0;
        float* aS = acc + ((size_t)bb * Mn + ns) * 8;
        float* aT = acc + ((size_t)bb * Mn + nt) * 8;
#pragma unroll
        for (int j = 0; j < 8; ++j) {
          atomicAdd(aS + j, cs[j]);
          atomicAdd(aT + j, cd[j]);
        }
      }
    }
  };

#if HAVE_ASYNC_LDS
  // -------- double-buffered async pipeline --------
  if (pr0 < nPairs) issueAsync(pr0, 0);
  int buf = 0;
  for (int pr = pr0; pr < nPairs; pr += stride) {
    const int prn = pr + stride;
    if (prn < nPairs) {
      issueAsync(prn, buf ^ 1);
      WAIT_ASYNC(4);  // in-order: oldest 4 (current buffer) are complete
    } else {
      WAIT_ASYNC(0);
    }
    computePair(pr, buf);
    buf ^= 1;
  }
#else
  // -------- sync staging fallback --------
  for (int pr = pr0; pr < nPairs; pr += stride) {
    const int e0 = 2 * pr;
    const int e1 = (e0 + 1 < E) ? e0 + 1 : e0;
    const int s0 = srcI[e0], t0 = dstI[e0];
    const int s1 = srcI[e1], t1 = dstI[e1];
    {
      const size_t ro = (size_t)(kh ? e1 : e0) * 64 + (size_t)lsel * 4;
      ((float4*)&Rb[0][w][0][0])[lane] = *(const float4*)(Rs + ro);
      ((float4*)&Rb[0][w][1][0])[lane] = *(const float4*)(Rd + ro);
    }
    {
      const int slice = lane >> 3;
      const int bb = lane & 7;
      const int node = (slice == 0) ? s0 : (slice == 1) ? t0
                                      : (slice == 2) ? s1 : t1;
      const float4* prow = (const float4*)(P + ((size_t)bb * Mn + node) * 8);
      float4* d4 = (float4*)&Pb[0][w][slice][bb][0];
      d4[0] = prow[0];
      d4[1] = prow[1];
    }
    const int prn = pr + stride;
    if (prn < nPairs) {  // prefetch next pair's rotor lines
      __builtin_prefetch(Rs + (size_t)(2 * prn) * 64, 0, 1);
      __builtin_prefetch(Rd + (size_t)(2 * prn) * 64, 0, 1);
    }
    __builtin_amdgcn_sched_barrier(0);
    computePair(pr, 0);
  }
#endif
}

// ---------------------------------------------------------------------------
// CG plumbing
// ---------------------------------------------------------------------------
__device__ __forceinline__ float block_reduce256(float v, float* sm) {
  const int t = threadIdx.x;
  sm[t] = v;
  __syncthreads();
  for (int o = 128; o > 0; o >>= 1) {
    if (t < o) sm[t] += sm[t + o];
    __syncthreads();
  }
  float r = sm[0];
  __syncthreads();
  return r;
}

__global__ void zero_acc_kernel(float* __restrict__ acc, int n4,
                                float* __restrict__ sc16) {
  const int stride = gridDim.x * blockDim.x;
  for (int i = blockIdx.x * blockDim.x + threadIdx.x; i < n4; i += stride)
    ((float4*)acc)[i] = make_float4(0.f, 0.f, 0.f, 0.f);
  if (blockIdx.x == 0 && threadIdx.x < 16) sc16[threadIdx.x] = 0.f;
}

__global__ void zero_state_kernel(float* __restrict__ rsold,
                                  int* __restrict__ flags) {
  if (threadIdx.x < 8) rsold[threadIdx.x] = 0.f;
  if (threadIdx.x == 0) flags[0] = 0;
}

// r = -lam*acc ; p = r ; rsold[b] += sum r^2
__global__ __launch_bounds__(256) void init_rp_kernel(
    const float* __restrict__ acc, float* __restrict__ r,
    float* __restrict__ p, float* __restrict__ rsold, int perBatch) {
  __shared__ float sm[256];
  const int b = blockIdx.y;
  const size_t off = (size_t)b * perBatch;
  float s = 0.f;
  const int stride = gridDim.x * blockDim.x;
  for (int i = blockIdx.x * blockDim.x + threadIdx.x; i < perBatch; i += stride) {
    const float rv = -LAM * acc[off + i];
    r[off + i] = rv;
    p[off + i] = rv;
    s += rv * rv;
  }
  const float tot = block_reduce256(s, sm);
  if (threadIdx.x == 0) atomicAdd(&rsold[b], tot);
}

// Ap = p + lam*acc (in place in acc) ; pAp[b] += p.Ap
__global__ __launch_bounds__(256) void combine_dot_kernel(
    const float* __restrict__ p, float* __restrict__ acc,
    float* __restrict__ pAp, int perBatch) {
  __shared__ float sm[256];
  const int b = blockIdx.y;
  const size_t off = (size_t)b * perBatch;
  float s = 0.f;
  const int stride = gridDim.x * blockDim.x;
  for (int i = blockIdx.x * blockDim.x + threadIdx.x; i < perBatch; i += stride) {
    const float pv = p[off + i];
    const float ap = pv + LAM * acc[off + i];
    acc[off + i] = ap;
    s += pv * ap;
  }
  const float tot = block_reduce256(s, sm);
  if (threadIdx.x == 0) atomicAdd(&pAp[b], tot);
}

__global__ void alpha_kernel(const float* __restrict__ rsold,
                             const float* __restrict__ pAp,
                             float* __restrict__ alpha,
                             const int* __restrict__ flags) {
  const int b = threadIdx.x;
  if (b < 8) alpha[b] = flags[0] ? 0.f : rsold[b] / (pAp[b] + 1e-12f);
}

// x += alpha p ; r -= alpha Ap ; rsnew[b] += sum r^2
__global__ __launch_bounds__(256) void update_xr_kernel(
    float* __restrict__ x, float* __restrict__ r, const float* __restrict__ p,
    const float* __restrict__ Ap, const float* __restrict__ alpha,
    float* __restrict__ rsnew, int perBatch) {
  __shared__ float sm[256];
  const int b = blockIdx.y;
  const float al = alpha[b];
  const size_t off = (size_t)b * perBatch;
  float s = 0.f;
  const int stride = gridDim.x * blockDim.x;
  for (int i = blockIdx.x * blockDim.x + threadIdx.x; i < perBatch; i += stride) {
    x[off + i] += al * p[off + i];
    const float rn = r[off + i] - al * Ap[off + i];
    r[off + i] = rn;
    s += rn * rn;
  }
  const float tot = block_reduce256(s, sm);
  if (threadIdx.x == 0) atomicAdd(&rsnew[b], tot);
}

// beta + convergence freeze (mean over the B per-batch sums, as in reference)
__global__ void scalars2_kernel(float* __restrict__ rsold,
                                const float* __restrict__ rsnew,
                                float* __restrict__ beta,
                                int* __restrict__ flags) {
  if (threadIdx.x == 0) {
    float tot = 0.f;
    for (int b = 0; b < 8; ++b) tot += rsnew[b];
    const int conv = (sqrtf(tot * 0.125f) < TOLV) ? 1 : 0;
    const int frozen = flags[0] | conv;
    for (int b = 0; b < 8; ++b) {
      beta[b] = frozen ? 0.f : rsnew[b] / (rsold[b] + 1e-12f);
      if (!frozen) rsold[b] = rsnew[b];
    }
    flags[0] = frozen;
  }
}

__global__ __launch_bounds__(256) void update_p_kernel(
    float* __restrict__ p, const float* __restrict__ r,
    const float* __restrict__ beta, const int* __restrict__ flags,
    int perBatch) {
  if (flags[0]) return;
  const int b = blockIdx.y;
  const float be = beta[b];
  const size_t off = (size_t)b * perBatch;
  const int stride = gridDim.x * blockDim.x;
  for (int i = blockIdx.x * blockDim.x + threadIdx.x; i < perBatch; i += stride)
    p[off + i] = r[off + i] + be * p[off + i];
}

extern "C" void kernel_launch(void* const* d_in, const int* in_sizes, int n_in,
                              void* d_out, int out_size, void* d_ws,
                              size_t ws_size, hipStream_t stream) {
  const float* c0 = (const float*)d_in[0];
  const int* srcI = (const int*)d_in[1];
  const int* dstI = (const int*)d_in[2];
  const float* Rs = (const float*)d_in[3];
  const float* Rd = (const float*)d_in[4];
  float* x = (float*)d_out;

  const int Ntot = in_sizes[0];  // B*M*d, B=d=8
  const int Mn = Ntot / 64;      // nodes
  const int perBatch = Mn * 8;   // M*d
  const int E = in_sizes[1];

  float* r = (float*)d_ws;
  float* p = r + Ntot;
  float* acc = p + Ntot;
  float* rsold = acc + Ntot;
  float* pAp = rsold + 8;  // pAp[8] + rsnew[8] contiguous (zeroed together)
  float* rsnew = pAp + 8;
  float* alpha = rsnew + 8;
  float* beta = alpha + 8;
  int* flags = (int*)(beta + 8);

  const int nPairs = (E + 1) >> 1;
  int edgeBlocks = (nPairs + WPB - 1) / WPB;
  if (edgeBlocks > 4096) edgeBlocks = 4096;
  const dim3 vgrid(128, 8);

  // x = c0
  (void)hipMemcpyAsync(x, c0, (size_t)Ntot * sizeof(float),
                       hipMemcpyDeviceToDevice, stream);
  zero_state_kernel<<<1, 64, 0, stream>>>(rsold, flags);

  // r = b - A x0 = -lam * L(c0)
  zero_acc_kernel<<<1024, 256, 0, stream>>>(acc, Ntot / 4, pAp);
  sheaf_edge_kernel<<<edgeBlocks, 256, 0, stream>>>(c0, srcI, dstI, Rs, Rd, acc,
                                                    Mn, E);
  init_rp_kernel<<<vgrid, 256, 0, stream>>>(acc, r, p, rsold, perBatch);

  for (int it = 0; it < NITERS; ++it) {
    zero_acc_kernel<<<1024, 256, 0, stream>>>(acc, Ntot / 4, pAp);
    sheaf_edge_kernel<<<edgeBlocks, 256, 0, stream>>>(p, srcI, dstI, Rs, Rd,
                                                      acc, Mn, E);
    combine_dot_kernel<<<vgrid, 256, 0, stream>>>(p, acc, pAp, perBatch);
    alpha_kernel<<<1, 32, 0, stream>>>(rsold, pAp, alpha, flags);
    update_xr_kernel<<<vgrid, 256, 0, stream>>>(x, r, p, acc, alpha, rsnew,
                                                perBatch);
    scalars2_kernel<<<1, 32, 0, stream>>>(rsold, rsnew, beta, flags);
    update_p_kernel<<<vgrid, 256, 0, stream>>>(p, r, beta, flags, perBatch);
  }
}